// HyperGraphCustomBipartiteDisenGATVAEV3CTRObjSameIdxHyperGraph_2276332666883
// MI455X (gfx1250) — compile-verified
//
#include <hip/hip_runtime.h>

typedef float v2f __attribute__((ext_vector_type(2)));
typedef float v8f __attribute__((ext_vector_type(8)));

#define WAVES_DENSE 4
#define HSTRIDE 68   // LDS row stride: mult of 4 -> float4 aligned; bank = (4*ln+c)%64 conflict-free

// ---------- helpers ----------

__device__ __forceinline__ float leaky(float x) { return x >= 0.0f ? x : 0.01f * x; }

// order-preserving float<->uint encoding for atomicMax on signed floats
__device__ __forceinline__ unsigned fenc(float f) {
    unsigned u = __float_as_uint(f);
    return (u & 0x80000000u) ? ~u : (u | 0x80000000u);
}
__device__ __forceinline__ float fdec(unsigned e) {
    return (e & 0x80000000u) ? __uint_as_float(e ^ 0x80000000u) : __uint_as_float(~e);
}

// One 16x16 f32 C tile of A[16x64] @ W[64x64] via 16 chained V_WMMA_F32_16X16X4_F32.
// Wp is the prepacked weight: element [(k*2+hh)*64 + n] = {W[(4k+2hh)*64+n], W[(4k+2hh+1)*64+n]}
// so each B fragment is a single contiguous b64 load (L2/WGP$ resident).
__device__ __forceinline__ v8f gemm16p(const v2f* a, const v2f* __restrict__ Wp,
                                       int nt, int hh, int ln) {
    const int n = nt * 16 + ln;
    v8f c = {0.f, 0.f, 0.f, 0.f, 0.f, 0.f, 0.f, 0.f};
#pragma unroll
    for (int k = 0; k < 16; ++k) {
        v2f b = Wp[(k * 2 + hh) * 64 + n];
        c = __builtin_amdgcn_wmma_f32_16x16x4_f32(false, a[k], false, b,
                                                  (short)0, c, false, false);
    }
    return c;
}

// Coalesced 16x64 tile copy global -> LDS (32 lanes x float4 = 512B/inst).
__device__ __forceinline__ void stage_tile(const float* __restrict__ src, float* dst, int lane) {
#pragma unroll
    for (int it = 0; it < 8; ++it) {
        int f = it * 128 + lane * 4;          // f = r*64 + c, rows contiguous
        int r = f >> 6, cc = f & 63;
        float4 v = *(const float4*)(src + f);
        *(float4*)(dst + r * HSTRIDE + cc) = v;
    }
}

// Extract per-lane A fragments from an LDS 16x64 tile (padded stride).
__device__ __forceinline__ void frag_from_lds(const float* buf, v2f* a, int hh, int ln) {
#pragma unroll
    for (int k = 0; k < 16; ++k) {
        const float* p = buf + ln * HSTRIDE + 4 * k + 2 * hh;
        a[k].x = p[0]; a[k].y = p[1];
    }
}

// ---------- init / prepack ----------

__global__ void init_m_kernel(unsigned* __restrict__ m, long long n) {
    long long t = (long long)blockIdx.x * blockDim.x + threadIdx.x;
    if (t < n) m[t] = 0x007FFFFFu;   // fenc(-inf)
}

// Pack 4 weight matrices into WMMA-B-fragment order (one float2 per lane read).
__global__ void prepack_W_kernel(const float* __restrict__ W0, const float* __restrict__ W1,
                                 const float* __restrict__ W2, const float* __restrict__ W3,
                                 v2f* __restrict__ Wp) {
    int t = blockIdx.x * blockDim.x + threadIdx.x;
    if (t >= 4 * 2048) return;
    int m   = t >> 11;
    int idx = t & 2047;          // (k*2+hh)*64 + n
    int n   = idx & 63;
    int kh  = idx >> 6;          // k*2 + hh
    int krow = kh * 2;           // 4k + 2hh
    const float* W = (m == 0) ? W0 : (m == 1) ? W1 : (m == 2) ? W2 : W3;
    v2f v;
    v.x = W[krow * 64 + n];
    v.y = W[(krow + 1) * 64 + n];
    Wp[t] = v;
}

// ---------- edge phase ----------

__global__ void edge_logits_kernel(const float* __restrict__ eu, const float* __restrict__ ei,
                                   const int* __restrict__ uidx, const int* __restrict__ iidx,
                                   float* __restrict__ logits,
                                   unsigned* __restrict__ m_u, unsigned* __restrict__ m_i,
                                   long long E) {
    long long g = (long long)blockIdx.x * blockDim.x + threadIdx.x;
    long long e = g >> 4;
    int j = (int)(g & 15);
    if (e >= E) return;
    int u = uidx[e], it = iidx[e];
    float4 a = *(const float4*)(eu + (long long)u * 64 + j * 4);
    float4 b = *(const float4*)(ei + (long long)it * 64 + j * 4);
    float s = a.x * b.x + a.y * b.y + a.z * b.z + a.w * b.w;
    s += __shfl_xor(s, 8);
    s += __shfl_xor(s, 4);
    s += __shfl_xor(s, 2);
    s += __shfl_xor(s, 1);
    if (j == 0) {
        float l = leaky(s);
        logits[e] = l;
        unsigned enc = fenc(l);
        atomicMax(m_i + it, enc);
        atomicMax(m_u + u, enc);
    }
}

__global__ void edge_expsum_kernel(const float* __restrict__ logits,
                                   const int* __restrict__ uidx, const int* __restrict__ iidx,
                                   const unsigned* __restrict__ m_u, const unsigned* __restrict__ m_i,
                                   float* __restrict__ s_u, float* __restrict__ s_i,
                                   long long E) {
    long long e = (long long)blockIdx.x * blockDim.x + threadIdx.x;
    if (e >= E) return;
    float l = logits[e];
    int it = iidx[e], u = uidx[e];
    atomicAdd(s_i + it, __expf(l - fdec(m_i[it])));
    atomicAdd(s_u + u,  __expf(l - fdec(m_u[u])));
}

__global__ void edge_agg_kernel(const float* __restrict__ eu, const float* __restrict__ ei,
                                const int* __restrict__ uidx, const int* __restrict__ iidx,
                                const float* __restrict__ logits,
                                const unsigned* __restrict__ m_u, const unsigned* __restrict__ m_i,
                                const float* __restrict__ s_u, const float* __restrict__ s_i,
                                float* __restrict__ user_agg, float* __restrict__ item_agg,
                                long long E) {
    long long g = (long long)blockIdx.x * blockDim.x + threadIdx.x;
    long long e = g >> 4;
    int j = (int)(g & 15);
    if (e >= E) return;
    int u = uidx[e], it = iidx[e];
    float l = logits[e];
    float ai = __expf(l - fdec(m_i[it])) / (s_i[it] + 1e-9f);
    float au = __expf(l - fdec(m_u[u]))  / (s_u[u] + 1e-9f);
    float4 a = *(const float4*)(eu + (long long)u * 64 + j * 4);
    float4 b = *(const float4*)(ei + (long long)it * 64 + j * 4);
    float* ia = item_agg + (long long)it * 64 + j * 4;
    atomicAdd(ia + 0, ai * a.x);
    atomicAdd(ia + 1, ai * a.y);
    atomicAdd(ia + 2, ai * a.z);
    atomicAdd(ia + 3, ai * a.w);
    float* ua = user_agg + (long long)u * 64 + j * 4;
    atomicAdd(ua + 0, au * b.x);
    atomicAdd(ua + 1, au * b.y);
    atomicAdd(ua + 2, au * b.z);
    atomicAdd(ua + 3, au * b.w);
}

// ---------- dense phase (WMMA, fused GCN -> VAE -> residual) ----------

__global__ void dense_user_kernel(const float* __restrict__ user_agg,
                                  const float* __restrict__ emb_user,
                                  const float* __restrict__ eps,
                                  const v2f* __restrict__ Wpg, const float* __restrict__ bg,
                                  const v2f* __restrict__ Wpm, const float* __restrict__ bm,
                                  const v2f* __restrict__ Wpl, const float* __restrict__ bl,
                                  const v2f* __restrict__ Wpd, const float* __restrict__ bd,
                                  float* __restrict__ out, int ntiles) {
    __shared__ float lds[WAVES_DENSE * 2 * 16 * HSTRIDE];
    int lane = threadIdx.x & 31;
    int wave = threadIdx.x >> 5;
    int tile = blockIdx.x * WAVES_DENSE + wave;
    if (tile >= ntiles) return;                 // wave-uniform: EXEC stays all-ones for WMMA
    float* hbuf = lds + wave * 2 * 16 * HSTRIDE;
    float* zbuf = hbuf + 16 * HSTRIDE;
    int hh = lane >> 4;
    int ln = lane & 15;
    long long base = (long long)tile * 1024;    // tile*16 rows * 64 cols

    // stage user_agg tile (coalesced b128) then fragment it
    stage_tile(user_agg + base, zbuf, lane);
    v2f a[16];
    frag_from_lds(zbuf, a, hh, ln);

    // GEMM 1: h = leaky(agg @ W_gcn + b_gcn) -> hbuf
#pragma unroll
    for (int nt = 0; nt < 4; ++nt) {
        v8f c = gemm16p(a, Wpg, nt, hh, ln);
        float bias = bg[nt * 16 + ln];
#pragma unroll
        for (int r = 0; r < 8; ++r)
            hbuf[(r + 8 * hh) * HSTRIDE + nt * 16 + ln] = leaky(c[r] + bias);
    }

    v2f a2[16];
    frag_from_lds(hbuf, a2, hh, ln);

    // stage eps tile (coalesced); z overwrites eps slot-for-slot (same lane, in order)
    stage_tile(eps + base, zbuf, lane);

    // GEMMs 2+3: mu / logvar -> reparameterize z -> zbuf
#pragma unroll
    for (int nt = 0; nt < 4; ++nt) {
        v8f cm = gemm16p(a2, Wpm, nt, hh, ln);
        v8f cl = gemm16p(a2, Wpl, nt, hh, ln);
        float bmu = bm[nt * 16 + ln], blv = bl[nt * 16 + ln];
#pragma unroll
        for (int r = 0; r < 8; ++r) {
            int slot = (r + 8 * hh) * HSTRIDE + nt * 16 + ln;
            float ev = zbuf[slot];
            zbuf[slot] = (cm[r] + bmu) + __expf(0.5f * (cl[r] + blv)) * ev;
        }
    }

    v2f a3[16];
    frag_from_lds(zbuf, a3, hh, ln);

    // GEMM 4: recon = z @ W_dec + b_dec; accumulate (h + recon) into zbuf
#pragma unroll
    for (int nt = 0; nt < 4; ++nt) {
        v8f cr = gemm16p(a3, Wpd, nt, hh, ln);
        float bde = bd[nt * 16 + ln];
#pragma unroll
        for (int r = 0; r < 8; ++r) {
            int slot = (r + 8 * hh) * HSTRIDE + nt * 16 + ln;
            zbuf[slot] = hbuf[slot] + (cr[r] + bde);
        }
    }

    // coalesced residual write: out = emb_user + (h + recon)
#pragma unroll
    for (int it = 0; it < 8; ++it) {
        int f = it * 128 + lane * 4;
        int r = f >> 6, cc = f & 63;
        float4 ev = *(const float4*)(emb_user + base + f);
        float4 sv = *(const float4*)(zbuf + r * HSTRIDE + cc);
        float4 ov;
        ov.x = ev.x + sv.x; ov.y = ev.y + sv.y;
        ov.z = ev.z + sv.z; ov.w = ev.w + sv.w;
        *(float4*)(out + base + f) = ov;
    }
}

__global__ void dense_item_kernel(const float* __restrict__ item_agg,
                                  const float* __restrict__ emb_item,
                                  const v2f* __restrict__ Wpg, const float* __restrict__ bg,
                                  float* __restrict__ out, int ntiles) {
    __shared__ float lds[WAVES_DENSE * 16 * HSTRIDE];
    int lane = threadIdx.x & 31;
    int wave = threadIdx.x >> 5;
    int tile = blockIdx.x * WAVES_DENSE + wave;
    if (tile >= ntiles) return;
    float* buf = lds + wave * 16 * HSTRIDE;
    int hh = lane >> 4, ln = lane & 15;
    long long base = (long long)tile * 1024;

    stage_tile(item_agg + base, buf, lane);
    v2f a[16];
    frag_from_lds(buf, a, hh, ln);

#pragma unroll
    for (int nt = 0; nt < 4; ++nt) {
        v8f c = gemm16p(a, Wpg, nt, hh, ln);
        float bias = bg[nt * 16 + ln];
#pragma unroll
        for (int r = 0; r < 8; ++r)
            buf[(r + 8 * hh) * HSTRIDE + nt * 16 + ln] = leaky(c[r] + bias);
    }

#pragma unroll
    for (int it = 0; it < 8; ++it) {
        int f = it * 128 + lane * 4;
        int r = f >> 6, cc = f & 63;
        float4 ev = *(const float4*)(emb_item + base + f);
        float4 sv = *(const float4*)(buf + r * HSTRIDE + cc);
        float4 ov;
        ov.x = ev.x + sv.x; ov.y = ev.y + sv.y;
        ov.z = ev.z + sv.z; ov.w = ev.w + sv.w;
        *(float4*)(out + base + f) = ov;
    }
}

// ---------- launch ----------

extern "C" void kernel_launch(void* const* d_in, const int* in_sizes, int n_in,
                              void* d_out, int out_size, void* d_ws, size_t ws_size,
                              hipStream_t stream) {
    (void)n_in; (void)out_size; (void)ws_size;
    const float* emb_user = (const float*)d_in[0];
    const float* emb_item = (const float*)d_in[1];
    const float* W_gcn    = (const float*)d_in[2];
    const float* b_gcn    = (const float*)d_in[3];
    const float* W_mu     = (const float*)d_in[4];
    const float* b_mu     = (const float*)d_in[5];
    const float* W_lv     = (const float*)d_in[6];
    const float* b_lv     = (const float*)d_in[7];
    const float* W_dec    = (const float*)d_in[8];
    const float* b_dec    = (const float*)d_in[9];
    const float* eps      = (const float*)d_in[10];
    const int*   u_idx    = (const int*)d_in[11];
    const int*   i_idx    = (const int*)d_in[12];
    float* out = (float*)d_out;

    const int D = 64;
    const long long U = in_sizes[0] / D;
    const long long I = in_sizes[1] / D;
    const long long E = in_sizes[11];

    // workspace layout
    float*    ws       = (float*)d_ws;
    float*    logits   = ws;                              // E
    unsigned* m_u      = (unsigned*)(ws + E);             // U
    unsigned* m_i      = m_u + U;                         // I
    float*    s_u      = (float*)(m_i + I);               // U
    float*    s_i      = s_u + U;                         // I
    float*    user_agg = s_i + I;                         // U*64
    float*    item_agg = user_agg + U * 64;               // I*64
    v2f*      Wp       = (v2f*)(item_agg + I * 64);       // 4 * 2048 float2

    // zero segment sums + aggregation buffers (contiguous region)
    hipMemsetAsync(s_u, 0, (size_t)((U + I) + (U + I) * 64) * sizeof(float), stream);
    long long nm = U + I;
    init_m_kernel<<<(unsigned)((nm + 255) / 256), 256, 0, stream>>>(m_u, nm);
    prepack_W_kernel<<<32, 256, 0, stream>>>(W_gcn, W_mu, W_lv, W_dec, Wp);

    // edge phase
    long long t16 = E * 16;
    edge_logits_kernel<<<(unsigned)((t16 + 255) / 256), 256, 0, stream>>>(
        emb_user, emb_item, u_idx, i_idx, logits, m_u, m_i, E);
    edge_expsum_kernel<<<(unsigned)((E + 255) / 256), 256, 0, stream>>>(
        logits, u_idx, i_idx, m_u, m_i, s_u, s_i, E);
    edge_agg_kernel<<<(unsigned)((t16 + 255) / 256), 256, 0, stream>>>(
        emb_user, emb_item, u_idx, i_idx, logits, m_u, m_i, s_u, s_i,
        user_agg, item_agg, E);

    // dense phase (fused WMMA chains), one wave per 16-row tile
    int ut  = (int)((U + 15) / 16);
    int itl = (int)((I + 15) / 16);
    dense_user_kernel<<<(ut + WAVES_DENSE - 1) / WAVES_DENSE, 32 * WAVES_DENSE, 0, stream>>>(
        user_agg, emb_user, eps, Wp, b_gcn, Wp + 2048, b_mu, Wp + 4096, b_lv, Wp + 6144, b_dec,
        out, ut);
    dense_item_kernel<<<(itl + WAVES_DENSE - 1) / WAVES_DENSE, 32 * WAVES_DENSE, 0, stream>>>(
        item_agg, emb_item, Wp, b_gcn, out + U * 64, itl);
}